// Net_84052509983312
// MI455X (gfx1250) — compile-verified
//
#include <hip/hip_runtime.h>

typedef __attribute__((ext_vector_type(2))) float v2f;
typedef __attribute__((ext_vector_type(8))) float v8f;

namespace {
constexpr int T_STEPS = 30;
constexpr int IN_F = 784;
constexpr int OUT_F = 10;
constexpr int KCHUNKS = IN_F / 4;              // 196 chunks of K=4
constexpr int KSPLIT = 4;                      // waves cooperating per 16-row tile
constexpr int KC_PER_WAVE = KCHUNKS / KSPLIT;  // 49
constexpr int TILES_PER_BLOCK = 2;
constexpr int THREADS = 32 * KSPLIT * TILES_PER_BLOCK;  // 256 = 8 waves
constexpr float BETA = 0.9375f;
constexpr float QS = 4096.0f;
}

__global__ __launch_bounds__(THREADS) void snn_leaky_wmma(
    const float* __restrict__ x, const float* __restrict__ W,
    float* __restrict__ out, int Btotal) {
  // B operand (Wq^T) pre-swizzled into V_WMMA_F32_16X16X4_F32 lane layout:
  // ldsB[kc*64 + lane*2 + c] = Wq[n][kc*4 + 2*(lane>>4) + c], n = lane&15
  __shared__ float ldsB[KCHUNKS * 64];                       // 50176 B
  __shared__ float ldsP[TILES_PER_BLOCK * (KSPLIT - 1) * 256];  // 6144 B partials

  const int tid = threadIdx.x;

  // Phase 1: quantize W (Q3.12, RNE like jnp.round) into LDS in B layout.
  for (int f = tid; f < KCHUNKS * 64; f += THREADS) {
    const int kc = f >> 6;
    const int r = f & 63;
    const int ln = r >> 1;
    const int comp = r & 1;
    const int n = ln & 15;
    const int kk = kc * 4 + ((ln >> 4) << 1) + comp;
    float v = 0.0f;
    if (n < OUT_F) {
      float q = rintf(W[n * IN_F + kk] * QS);
      q = fminf(fmaxf(q, -32768.0f), 32767.0f);
      v = q * (1.0f / QS);
    }
    ldsB[f] = v;
  }
  __syncthreads();

  const int lane = tid & 31;
  const int wv = tid >> 5;
  const int kwave = wv & (KSPLIT - 1);  // K-slice owner within tile group
  const int tloc = wv / KSPLIT;         // tile within block
  const int tile = blockIdx.x * TILES_PER_BLOCK + tloc;
  const int b0 = tile * 16;
  const bool active = (b0 + 16) <= Btotal;

  const int m = lane & 15;    // A-operand row (M)
  const int khalf = lane >> 4;

  const float* xbase =
      x + (size_t)(b0 + m) * IN_F + kwave * (KC_PER_WAVE * 4) + khalf * 2;
  const size_t xstep = (size_t)Btotal * IN_F;
  const int kcg0 = kwave * KC_PER_WAVE;

  const int slot = (kwave == 0) ? 0 : (tloc * (KSPLIT - 1) + kwave - 1);
  float* pslot = &ldsP[slot * 256 + lane];
  const float* pbase = &ldsP[tloc * (KSPLIT - 1) * 256 + lane];

  v8f mem = {0.f, 0.f, 0.f, 0.f, 0.f, 0.f, 0.f, 0.f};
  const size_t memOutBase = (size_t)T_STEPS * Btotal * OUT_F;

  for (int t = 0; t < T_STEPS; ++t) {
    v8f cur = {0.f, 0.f, 0.f, 0.f, 0.f, 0.f, 0.f, 0.f};
    if (active) {
      const float* xt = xbase + (size_t)t * xstep;
      v8f acc0 = {0.f, 0.f, 0.f, 0.f, 0.f, 0.f, 0.f, 0.f};
      v8f acc1 = {0.f, 0.f, 0.f, 0.f, 0.f, 0.f, 0.f, 0.f};
      int i = 0;
#pragma unroll 4
      for (; i + 1 < KC_PER_WAVE; i += 2) {
        v2f a0 = *(const v2f*)(xt + i * 4);
        v2f bb0 = *(const v2f*)&ldsB[(kcg0 + i) * 64 + lane * 2];
        acc0 = __builtin_amdgcn_wmma_f32_16x16x4_f32(false, a0, false, bb0,
                                                     (short)0, acc0, false, false);
        v2f a1 = *(const v2f*)(xt + (i + 1) * 4);
        v2f bb1 = *(const v2f*)&ldsB[(kcg0 + i + 1) * 64 + lane * 2];
        acc1 = __builtin_amdgcn_wmma_f32_16x16x4_f32(false, a1, false, bb1,
                                                     (short)0, acc1, false, false);
      }
      {  // tail: KC_PER_WAVE = 49 is odd
        v2f a0 = *(const v2f*)(xt + i * 4);
        v2f bb0 = *(const v2f*)&ldsB[(kcg0 + i) * 64 + lane * 2];
        acc0 = __builtin_amdgcn_wmma_f32_16x16x4_f32(false, a0, false, bb0,
                                                     (short)0, acc0, false, false);
      }
      cur = acc0 + acc1;
    }

    // Cross-wave K reduction through LDS, then scan in wave 0 of each group.
    if (kwave != 0) {
#pragma unroll
      for (int r = 0; r < 8; ++r) pslot[r * 32] = cur[r];
    }
    __syncthreads();
    if (kwave == 0 && active) {
      const int n = lane & 15;               // D layout: N = lane&15
      const int mbase = b0 + 8 * khalf;      // M = r + 8*(lane>>4)
#pragma unroll
      for (int r = 0; r < 8; ++r) {
        float c = cur[r] + pbase[0 * 256 + r * 32] + pbase[1 * 256 + r * 32] +
                  pbase[2 * 256 + r * 32];
        float mo = mem[r];
        float reset = (mo > 1.0f) ? 1.0f : 0.0f;   // reset from previous mem
        float mn = BETA * mo + c - reset;          // subtract-reset Leaky
        mem[r] = mn;
        if (n < OUT_F) {
          size_t idx = ((size_t)t * Btotal + (size_t)(mbase + r)) * OUT_F + n;
          out[idx] = (mn > 1.0f) ? 1.0f : 0.0f;    // spk_rec
          out[memOutBase + idx] = mn;              // mem_rec
        }
      }
    }
    __syncthreads();  // protect ldsP before next timestep
  }
}

extern "C" void kernel_launch(void* const* d_in, const int* in_sizes, int n_in,
                              void* d_out, int out_size, void* d_ws,
                              size_t ws_size, hipStream_t stream) {
  (void)n_in; (void)out_size; (void)d_ws; (void)ws_size;
  const float* x = (const float*)d_in[0];
  const float* W = (const float*)d_in[1];
  float* out = (float*)d_out;

  const int Btotal = in_sizes[0] / (T_STEPS * IN_F);  // 8192
  const int tiles = (Btotal + 15) / 16;               // 512
  const int blocks = (tiles + TILES_PER_BLOCK - 1) / TILES_PER_BLOCK;  // 256

  snn_leaky_wmma<<<blocks, THREADS, 0, stream>>>(x, W, out, Btotal);
}